// MorphPool3D_7619271983644
// MI455X (gfx1250) — compile-verified
//
#include <hip/hip_runtime.h>
#include <hip/hip_bf16.h>
#include <stdint.h>

// Geometry (fixed by the reference): B=4, C=1, D=H=W=160.
#define DGRID 160
#define WPR 5                                   // 160 bits / 32 = 5 real words per row
#define ROW_STRIDE 8                            // padded to 32B for b128 async staging
#define ROWS_PER_BATCH (DGRID*DGRID)            // 25600
#define PAD_WORDS_PER_BATCH (ROWS_PER_BATCH*ROW_STRIDE)  // 204800
#define REAL_WORDS_TOTAL (4*ROWS_PER_BATCH*WPR)          // 512000
#define TOTAL_ELEMS (4*DGRID*DGRID*DGRID)                // 16,384,000

#define TILE_Y 32
#define HALO_ROWS (TILE_Y+2)                    // 34
#define LDS_WORDS (3*HALO_ROWS*ROW_STRIDE)      // 816 words = 3264 B

// ---------------------------------------------------------------------------
// Stage 1: binarize + bitpack via wave32 ballot. One wave -> one u32 word.
// pred = (aux<0) || (aux==0 && inp!=0)   (== "t != 0" in the reference)
// ---------------------------------------------------------------------------
__global__ __launch_bounds__(256)
void pack_kernel(const float* __restrict__ inp, const float* __restrict__ aux,
                 uint32_t* __restrict__ dst) {
  const int gtid = blockIdx.x * blockDim.x + threadIdx.x;
  const int wave = gtid >> 5;                  // one packed word per wave
  const int lane = threadIdx.x & 31;
  if (wave >= REAL_WORDS_TOTAL) return;
  const int row = wave / WPR;                  // ((b*160+z)*160+y)
  const int w   = wave - row * WPR;            // 0..4
  const size_t e = (size_t)row * DGRID + w * 32 + lane;  // linear element index
  const float a = aux[e];
  const float x = inp[e];
  const bool pred = (a < 0.0f) || (a == 0.0f && x != 0.0f);
  const uint32_t mask = __builtin_amdgcn_ballot_w32(pred);
  if (lane == 0) dst[(size_t)row * ROW_STRIDE + w] = mask;
}

// ---------------------------------------------------------------------------
// Stage 2: one morphology pass (one 9-offset plane), bitpacked, LDS-staged
// through the CDNA5 async global->LDS path (ASYNCcnt).
// ---------------------------------------------------------------------------
// Structuring-element offsets (di along D/z, dj along H/y, dk along W/bits),
// matching _make_offsets in the reference. o enumerates 9 combos of (a,t).
template <int P>
__device__ __forceinline__ void get_off(int o, int& di, int& dj, int& dk) {
  const int a = o / 3 - 1;
  const int t = o % 3 - 1;
  if      (P == 0) { di = a; dj = t;  dk = 0;  }
  else if (P == 1) { di = a; dj = 0;  dk = t;  }
  else if (P == 2) { di = 0; dj = a;  dk = t;  }
  else if (P == 3) { di = a; dj = t;  dk = t;  }
  else if (P == 4) { di = a; dj = t;  dk = -t; }
  else if (P == 5) { di = t; dj = a;  dk = t;  }
  else if (P == 6) { di = t; dj = a;  dk = -t; }
  else if (P == 7) { di = t; dj = t;  dk = a;  }
  else             { di = t; dj = -t; dk = a;  }
}

__device__ __forceinline__ uint32_t ldw(const uint32_t* sh, int zi, int yi, int w) {
  // Words outside [0,5) (incl. the 3 padding words) act as pad-0.
  return (w < 0 || w >= WPR) ? 0u : sh[(zi * HALO_ROWS + yi) * ROW_STRIDE + w];
}

// out[k] = in[k+dk]: bit t of word w is element k = w*32+t, so dk=+1 is a
// funnel shift-right with carry from the next word, dk=-1 shift-left.
__device__ __forceinline__ uint32_t rowsh(const uint32_t* sh, int zi, int yi,
                                          int w, int dk) {
  const uint32_t c = ldw(sh, zi, yi, w);
  if (dk == 0) return c;
  if (dk > 0)  return (c >> 1) | (ldw(sh, zi, yi, w + 1) << 31);
  return (c << 1) | (ldw(sh, zi, yi, w - 1) >> 31);
}

template <int P>
__global__ __launch_bounds__(160)
void morph_pass(const uint32_t* __restrict__ src, uint32_t* __restrict__ dst,
                int chain) {
  __shared__ uint32_t sh[LDS_WORDS];
  const int tile = blockIdx.x;   // 0..4 (exact y tiles of 32)
  const int z    = blockIdx.y;   // 0..159
  const int b    = blockIdx.z;   // 0..3
  const int t    = threadIdx.x;  // 0..159
  const int y0   = tile * TILE_Y;

  // Zero-fill LDS so out-of-range halo rows behave as pad-0.
  for (int s = t; s < LDS_WORDS; s += 160) sh[s] = 0u;
  __syncthreads();

  // Stage rows (z-1..z+1) x (y0-1..y0+32): one b128 (words 0-3, 16B aligned)
  // + one b32 (word 4) per row. Rows are 32B aligned by construction.
  const uint32_t* batch_src = src + (size_t)b * PAD_WORDS_PER_BATCH;
  if (t < 3 * HALO_ROWS) {
    const int zi = t / HALO_ROWS;
    const int yi = t - zi * HALO_ROWS;
    const int zg = z - 1 + zi;
    const int yg = y0 - 1 + yi;
    if ((unsigned)zg < (unsigned)DGRID && (unsigned)yg < (unsigned)DGRID) {
      const uint32_t* gp = batch_src + ((size_t)zg * DGRID + yg) * ROW_STRIDE;
      const uint32_t la = (uint32_t)(uintptr_t)(&sh[(zi * HALO_ROWS + yi) * ROW_STRIDE]);
      asm volatile("global_load_async_to_lds_b128 %0, %1, off"
                   :: "v"(la), "v"(gp) : "memory");
      asm volatile("global_load_async_to_lds_b32 %0, %1, off offset:16"
                   :: "v"(la + 16), "v"(gp) : "memory");
    }
  }
  asm volatile("s_wait_asynccnt 0" ::: "memory");
  __syncthreads();

  const int yl = t / WPR;        // 0..31
  const int w  = t - yl * WPR;   // 0..4
  const int y  = y0 + yl;        // always < 160 (exact tiling)

  // even batch: chain0 = dilate(OR), chain1 = erode(AND); odd batch: swapped.
  const bool mx = ((b & 1) == 0) ? (chain == 0) : (chain != 0);

  uint32_t acc = 0;
  bool first = true;
#pragma unroll
  for (int o = 0; o < 9; ++o) {
    int di, dj, dk;
    get_off<P>(o, di, dj, dk);
    const uint32_t v = rowsh(sh, 1 + di, yl + 1 + dj, w, dk);
    if (first) { acc = v; first = false; }
    else       { acc = mx ? (acc | v) : (acc & v); }
  }
  dst[(size_t)b * PAD_WORDS_PER_BATCH + ((size_t)z * DGRID + y) * ROW_STRIDE + w] = acc;
}

// ---------------------------------------------------------------------------
// Stage 3: unpack bits -> float32 output.
// ---------------------------------------------------------------------------
__global__ __launch_bounds__(256)
void unpack_kernel(const uint32_t* __restrict__ bits, float* __restrict__ out,
                   int n) {
  const int i = blockIdx.x * blockDim.x + threadIdx.x;
  if (i >= n) return;
  const int row = i / DGRID;
  const int k   = i - row * DGRID;
  const uint32_t wv = bits[(size_t)row * ROW_STRIDE + (k >> 5)];
  out[i] = ((wv >> (k & 31)) & 1u) ? 1.0f : 0.0f;
}

// ---------------------------------------------------------------------------
static void launch_pass(int p, const uint32_t* s, uint32_t* d, int chain,
                        hipStream_t st) {
  const dim3 g(DGRID / TILE_Y, DGRID, 4);
  switch (p) {
    case 0: morph_pass<0><<<g, 160, 0, st>>>(s, d, chain); break;
    case 1: morph_pass<1><<<g, 160, 0, st>>>(s, d, chain); break;
    case 2: morph_pass<2><<<g, 160, 0, st>>>(s, d, chain); break;
    case 3: morph_pass<3><<<g, 160, 0, st>>>(s, d, chain); break;
    case 4: morph_pass<4><<<g, 160, 0, st>>>(s, d, chain); break;
    case 5: morph_pass<5><<<g, 160, 0, st>>>(s, d, chain); break;
    case 6: morph_pass<6><<<g, 160, 0, st>>>(s, d, chain); break;
    case 7: morph_pass<7><<<g, 160, 0, st>>>(s, d, chain); break;
    default: morph_pass<8><<<g, 160, 0, st>>>(s, d, chain); break;
  }
}

extern "C" void kernel_launch(void* const* d_in, const int* in_sizes, int n_in,
                              void* d_out, int out_size, void* d_ws, size_t ws_size,
                              hipStream_t stream) {
  (void)in_sizes; (void)n_in; (void)ws_size;
  const float* inp = (const float*)d_in[0];
  const float* aux = (const float*)d_in[1];
  float* out = (float*)d_out;

  uint32_t* bufA = (uint32_t*)d_ws;
  uint32_t* bufB = bufA + 4 * PAD_WORDS_PER_BATCH;  // 2 x 3.28 MB ping-pong in L2

  pack_kernel<<<(TOTAL_ELEMS + 255) / 256, 256, 0, stream>>>(inp, aux, bufA);

  // 2 chains x 9 planes = 18 passes; even count -> result lands back in bufA.
  const uint32_t* src = bufA;
  uint32_t* dst = bufB;
  for (int chain = 0; chain < 2; ++chain) {
    for (int p = 0; p < 9; ++p) {
      launch_pass(p, src, dst, chain, stream);
      const uint32_t* tmp = dst;
      dst = (uint32_t*)src;
      src = tmp;
    }
  }

  const int n = out_size;  // 16,384,000
  unpack_kernel<<<(n + 255) / 256, 256, 0, stream>>>(src, out, n);
}